// MultiScaleEdgeBuilder_14199161880846
// MI455X (gfx1250) — compile-verified
//
#include <hip/hip_runtime.h>

// CDNA5 / gfx1250, wave32. WMMA f32 16x16x4 used for per-graph Gram matrices.
// All output stores are non-temporal: 351 MB of write-once data > 192 MB L2.

typedef __attribute__((ext_vector_type(2))) float v2f;
typedef __attribute__((ext_vector_type(4))) float v4f;   // clang native vector
typedef __attribute__((ext_vector_type(8))) float v8f;

#define NPER       256
#define NGRAPH     64
#define EPG        (NPER * (NPER - 1))        // 65280 edges per graph
#define ETOT       ((long long)NGRAPH * EPG)  // 4,177,920
#define ROWS_BLK   32                         // rows of d2 tile kept in LDS
#define BLK_PER_G  (NPER / ROWS_BLK)          // 8

// ---------------------------------------------------------------------------
// Kernel 1: zero-fill edge_attr region (16E floats at offset 2E) with
// non-temporal 128-bit stores; this is the dominant bandwidth cost (~267 MB).
// ---------------------------------------------------------------------------
__global__ __launch_bounds__(256) void edge_attr_zero_kernel(float* __restrict__ out) {
    const long long nvec = (16LL * ETOT) / 4;            // 128-bit store count
    v4f* __restrict__ p = reinterpret_cast<v4f*>(out + 2LL * ETOT);
    const v4f z = {0.f, 0.f, 0.f, 0.f};
    long long i = (long long)blockIdx.x * blockDim.x + threadIdx.x;
    const long long stride = (long long)gridDim.x * blockDim.x;
    for (; i < nvec; i += stride) {
        __builtin_nontemporal_store(z, &p[i]);
    }
}

// ---------------------------------------------------------------------------
// Kernel 2: per (graph, 32-row block): WMMA Gram tiles -> d2 in LDS -> stream
// out edge_index (row/col) and rbf = exp(-d2/cut^2), all coalesced + NT.
// ---------------------------------------------------------------------------
__global__ __launch_bounds__(256) void edge_rbf_kernel(const float* __restrict__ pos,
                                                       float* __restrict__ out) {
    __shared__ float4 sPos[NPER];                  // x, y, z, |x|^2
    __shared__ float  sD2[ROWS_BLK * NPER];        // 32 KB d2 tile

    const int tid  = threadIdx.x;
    const int g    = blockIdx.x >> 3;              // graph id
    const int r0   = (blockIdx.x & 7) * ROWS_BLK;  // first row of this block
    const int goff = g * NPER;                     // global node offset

    // Load this graph's 256 positions (768 floats) into LDS, precompute |x|^2.
    {
        const float* p = pos + (size_t)goff * 3 + (size_t)tid * 3;
        float x = p[0], y = p[1], z = p[2];
        sPos[tid] = make_float4(x, y, z, x * x + y * y + z * z);
    }
    __syncthreads();

    const int  wave = tid >> 5;
    const int  lane = tid & 31;
    const int  ln   = lane & 15;
    const bool hi   = lane >= 16;

    // 32-row block = 2 row-tiles x 16 col-tiles = 32 WMMA tiles; 8 waves x 4.
    for (int q = 0; q < 4; ++q) {
        const int t  = wave * 4 + q;
        const int i0 = r0 + (t >> 4) * 16;         // tile row origin (within graph)
        const int j0 = (t & 15) * 16;              // tile col origin

        // A: 16x4 f32 (rows i0..i0+15, K = x,y,z,0)
        //   lanes 0-15 : VGPR0=K0(x), VGPR1=K1(y);  lanes 16-31: VGPR0=K2(z), VGPR1=K3(0)
        // B: 4x16 f32 (cols j0..j0+15), same K striping.
        const float4 pa = sPos[i0 + ln];
        const float4 pb = sPos[j0 + ln];
        v2f a, b;
        a[0] = hi ? pa.z : pa.x;  a[1] = hi ? 0.f : pa.y;
        b[0] = hi ? pb.z : pb.x;  b[1] = hi ? 0.f : pb.y;

        v8f c = {};
        c = __builtin_amdgcn_wmma_f32_16x16x4_f32(false, a, false, b,
                                                  (short)0, c, false, false);

        // C/D layout: vreg v, lanes 0-15 -> row v, lanes 16-31 -> row v+8; col = ln.
        const int   mbase = (t >> 4) * 16 + (hi ? 8 : 0);  // row within 32-row block
        const int   n     = j0 + ln;                       // col (node in graph)
        const float rj    = sPos[n].w;
#pragma unroll
        for (int v = 0; v < 8; ++v) {
            const int   m  = mbase + v;
            const float ri = sPos[r0 + m].w;
            sD2[m * NPER + n] = ri + rj - 2.0f * c[v];
        }
    }
    __syncthreads();

    // First edge (global) of this block; each row has NPER-1 edges.
    const long long ebase  = (long long)g * EPG + (long long)r0 * (NPER - 1);
    float* __restrict__ rowOut = out;                    // [E]
    float* __restrict__ colOut = out + ETOT;             // [E]
    float* __restrict__ rbfOut = out + 18LL * ETOT;      // [E,3]
    const int nEdge = ROWS_BLK * (NPER - 1);             // 8160

    // edge_index: one float per thread per iteration, fully coalesced, NT.
    for (int e = tid; e < nEdge; e += 256) {
        const int il = e / (NPER - 1);
        const int r  = e - il * (NPER - 1);
        const int ig = r0 + il;
        const int j  = r + (r >= ig ? 1 : 0);            // skip diagonal
        const long long eg = ebase + e;
        __builtin_nontemporal_store((float)(goff + ig), &rowOut[eg]);
        __builtin_nontemporal_store((float)(goff + j),  &colOut[eg]);
    }

    // rbf: one thread per output float (3 per edge), fully coalesced, NT.
    const float invc2[3] = {1.0f / 16.0f, 1.0f / 64.0f, 1.0f / 144.0f};
    float* __restrict__ rbfBlk = rbfOut + ebase * 3;
    const int nF = nEdge * 3;
    for (int tt = tid; tt < nF; tt += 256) {
        const int e  = tt / 3;
        const int k  = tt - e * 3;
        const int il = e / (NPER - 1);
        const int r  = e - il * (NPER - 1);
        const int ig = r0 + il;
        const int j  = r + (r >= ig ? 1 : 0);
        const float d2 = fmaxf(sD2[il * NPER + j], 0.0f);
        __builtin_nontemporal_store(__expf(-d2 * invc2[k]), &rbfBlk[tt]);
    }
}

// ---------------------------------------------------------------------------
extern "C" void kernel_launch(void* const* d_in, const int* in_sizes, int n_in,
                              void* d_out, int out_size, void* d_ws, size_t ws_size,
                              hipStream_t stream) {
    (void)in_sizes; (void)n_in; (void)d_ws; (void)ws_size; (void)out_size;
    const float* pos = (const float*)d_in[0];   // [N,3] float32
    float* out = (float*)d_out;                 // [2E | 16E | 3E] as float

    edge_attr_zero_kernel<<<dim3(8192), dim3(256), 0, stream>>>(out);
    edge_rbf_kernel<<<dim3(NGRAPH * BLK_PER_G), dim3(256), 0, stream>>>(pos, out);
}